// GCN_52012053955018
// MI455X (gfx1250) — compile-verified
//
#include <hip/hip_runtime.h>
#include <hip/hip_bf16.h>
#include <math.h>

typedef __attribute__((ext_vector_type(2))) float v2f;
typedef __attribute__((ext_vector_type(8))) float v8f;

#define NFEAT 512
#define NHID 8
#define NCLASS 16

// ---------------------------------------------------------------------------
// Kernel 1: fused dense projections with fp32 WMMA (V_WMMA_F32_16X16X4_F32).
// One wave per 16-node tile. Single pass over x feeds three WMMA chains:
//   acc1 = x @ W1 ; accV = x @ V ; acc2 = (x*x) @ (V*V)
// Hidden dim 8 padded to WMMA N=16 with zero B-columns.
// A fragment (16x4 f32): lanes 0-15 hold K={k,k+1}, lanes 16-31 K={k+2,k+3}.
// B fragment (4x16 f32): v0 = rows {k, k+2}, v1 = rows {k+1, k+3} across lanes.
// ---------------------------------------------------------------------------
__global__ __launch_bounds__(256) void k_dense_wmma(
    const float* __restrict__ x, const float* __restrict__ W1,
    const float* __restrict__ V, const float* __restrict__ gamma,
    const float* __restrict__ beta,
    float* __restrict__ xw1_ws, float* __restrict__ xr_ws, int N)
{
    int wave = threadIdx.x >> 5;
    int lane = threadIdx.x & 31;
    int tile = blockIdx.x * (blockDim.x >> 5) + wave;
    int tiles = (N + 15) >> 4;
    if (tile >= tiles) return;            // wave-uniform exit: EXEC stays all-1 for WMMA

    int half  = lane >> 4;                // 0 or 1
    int khalf = half * 2;                 // K sub-offset: 0 or 2
    int rloc  = lane & 15;
    int row   = tile * 16 + rloc;
    int rowc  = row < N ? row : N - 1;    // clamp (N=100000 is 16-divisible anyway)
    int n     = lane & 15;                // B/D column this lane owns
    int nc    = n & 7;
    float bmask = (n < NHID) ? 1.0f : 0.0f;

    const float* xrow = x  + (size_t)rowc * NFEAT + khalf;
    const float* w1p  = W1 + khalf * NHID + nc;   // W1 is [512][8] row-major
    const float* vp   = V  + khalf * NHID + nc;

    v8f acc1 = {};
    v8f accV = {};
    v8f acc2 = {};

    for (int k = 0; k < NFEAT; k += 4) {
        v2f a  = *(const v2f*)(xrow + k);          // x[r][k+khalf .. +1], 8B aligned
        v2f a2 = a * a;
        v2f bw, bv, bvv;
        bw.x = w1p[(k    ) * NHID] * bmask;        // W1[k+khalf  ][n]
        bw.y = w1p[(k + 1) * NHID] * bmask;        // W1[k+khalf+1][n]
        bv.x = vp [(k    ) * NHID] * bmask;
        bv.y = vp [(k + 1) * NHID] * bmask;
        bvv  = bv * bv;
        acc1 = __builtin_amdgcn_wmma_f32_16x16x4_f32(false, a,  false, bw,  (short)0, acc1, false, false);
        accV = __builtin_amdgcn_wmma_f32_16x16x4_f32(false, a,  false, bv,  (short)0, accV, false, false);
        acc2 = __builtin_amdgcn_wmma_f32_16x16x4_f32(false, a2, false, bvv, (short)0, acc2, false, false);
    }

    // D layout (f32 16x16): VGPR j -> row j (lanes 0-15) / row j+8 (lanes 16-31), col = lane%16
    int mbase = tile * 16 + 8 * half;
    float g  = gamma[nc];
    float bb = beta[nc];
    if (n < NHID) {
        #pragma unroll
        for (int j = 0; j < 8; ++j) {
            int m = mbase + j;
            if (m < N) {
                float xw   = acc1[j];
                float xv   = accV[j];
                float x2v2 = acc2[j];
                float xr   = 0.5f * (xv * xv - x2v2);      // FM bi-interaction
                xr = fmaxf(xr, 0.0f);                      // relu
                xr = g * xr + bb;                          // BN eval affine
                xw1_ws[(size_t)m * NHID + n] = xw;
                xr_ws [(size_t)m * NHID + n] = xr;
            }
        }
    }
}

// ---------------------------------------------------------------------------
__global__ void k_zero(float* __restrict__ p, size_t n) {
    size_t i = (size_t)blockIdx.x * blockDim.x + threadIdx.x;
    if (i < n) p[i] = 0.0f;
}

// SpMM (COO) via atomics, 8 features/edge: dst[row] += val * src[col]
__global__ void k_spmm8(const int* __restrict__ rows, const int* __restrict__ cols,
                        const float* __restrict__ vals, const float* __restrict__ src,
                        float* __restrict__ dst, int E)
{
    long long t = (long long)blockIdx.x * blockDim.x + threadIdx.x;
    int e = (int)(t >> 3);
    int f = (int)(t & 7);
    if (e >= E) return;
    float v = vals[e];
    int c = cols[e];
    int r = rows[e];
    atomicAdd(&dst[(size_t)r * 8 + f], v * src[(size_t)c * 8 + f]);
}

// SpMM (COO) via atomics, 16 features/edge
__global__ void k_spmm16(const int* __restrict__ rows, const int* __restrict__ cols,
                         const float* __restrict__ vals, const float* __restrict__ src,
                         float* __restrict__ dst, int E)
{
    long long t = (long long)blockIdx.x * blockDim.x + threadIdx.x;
    int e = (int)(t >> 4);
    int f = (int)(t & 15);
    if (e >= E) return;
    float v = vals[e];
    int c = cols[e];
    int r = rows[e];
    atomicAdd(&dst[(size_t)r * 16 + f], v * src[(size_t)c * 16 + f]);
}

// h = 0.5*(relu(spmm1 + b1) + x_right);  hw2 = h @ W2   ([8]x[8,16] per node)
__global__ void k_hw2(const float* __restrict__ acc1, const float* __restrict__ xr,
                      const float* __restrict__ b1, const float* __restrict__ W2,
                      float* __restrict__ hw2, int N)
{
    int nidx = blockIdx.x * blockDim.x + threadIdx.x;
    if (nidx >= N) return;
    float h[NHID];
    #pragma unroll
    for (int j = 0; j < NHID; ++j) {
        float t = acc1[(size_t)nidx * NHID + j] + b1[j];
        t = fmaxf(t, 0.0f);
        h[j] = 0.5f * (t + xr[(size_t)nidx * NHID + j]);
    }
    #pragma unroll
    for (int c = 0; c < NCLASS; ++c) {
        float s = 0.0f;
        #pragma unroll
        for (int j = 0; j < NHID; ++j) s += h[j] * W2[j * NCLASS + c];
        hw2[(size_t)nidx * NCLASS + c] = s;
    }
}

// out = log_softmax(spmm2 + b2) over 16 classes
__global__ void k_lsm(const float* __restrict__ acc2, const float* __restrict__ b2,
                      float* __restrict__ out, int N)
{
    int nidx = blockIdx.x * blockDim.x + threadIdx.x;
    if (nidx >= N) return;
    float z[NCLASS];
    float m = -INFINITY;
    #pragma unroll
    for (int c = 0; c < NCLASS; ++c) {
        z[c] = acc2[(size_t)nidx * NCLASS + c] + b2[c];
        m = fmaxf(m, z[c]);
    }
    float s = 0.0f;
    #pragma unroll
    for (int c = 0; c < NCLASS; ++c) s += expf(z[c] - m);
    float ls = logf(s);
    #pragma unroll
    for (int c = 0; c < NCLASS; ++c)
        out[(size_t)nidx * NCLASS + c] = z[c] - m - ls;
}

// ---------------------------------------------------------------------------
extern "C" void kernel_launch(void* const* d_in, const int* in_sizes, int n_in,
                              void* d_out, int out_size, void* d_ws, size_t ws_size,
                              hipStream_t stream)
{
    (void)n_in; (void)out_size; (void)ws_size;
    const float* x     = (const float*)d_in[0];
    const int*   rows  = (const int*)  d_in[1];
    const int*   cols  = (const int*)  d_in[2];
    const float* vals  = (const float*)d_in[3];
    const float* W1    = (const float*)d_in[4];
    const float* b1    = (const float*)d_in[5];
    const float* W2    = (const float*)d_in[6];
    const float* b2    = (const float*)d_in[7];
    const float* V     = (const float*)d_in[8];
    const float* gamma = (const float*)d_in[9];
    const float* beta  = (const float*)d_in[10];
    float* out = (float*)d_out;
    float* ws  = (float*)d_ws;

    const int N = in_sizes[0] / NFEAT;     // 100000
    const int E = in_sizes[1];             // 3200000

    // workspace layout (floats): xw1[N*8] | xr[N*8] | hw2[N*16] | acc1[N*8] | acc2[N*16]
    float* xw1_ws  = ws;
    float* xr_ws   = ws + (size_t)N * 8;
    float* hw2_ws  = ws + (size_t)N * 16;
    float* acc1_ws = ws + (size_t)N * 32;
    float* acc2_ws = ws + (size_t)N * 40;  // total N*56 floats = 22.4 MB

    // zero the two atomic accumulators (contiguous region, N*24 floats)
    size_t zn = (size_t)N * 24;
    k_zero<<<(unsigned)((zn + 255) / 256), 256, 0, stream>>>(acc1_ws, zn);

    // fused dense projections (WMMA)
    int tiles = (N + 15) / 16;
    k_dense_wmma<<<(tiles + 7) / 8, 256, 0, stream>>>(x, W1, V, gamma, beta,
                                                      xw1_ws, xr_ws, N);

    // SpMM1: acc1 = A @ (x@W1)
    long long t8 = (long long)E * 8;
    k_spmm8<<<(unsigned)((t8 + 255) / 256), 256, 0, stream>>>(rows, cols, vals,
                                                              xw1_ws, acc1_ws, E);

    // h + h@W2
    k_hw2<<<(N + 255) / 256, 256, 0, stream>>>(acc1_ws, xr_ws, b1, W2, hw2_ws, N);

    // SpMM2: acc2 = A @ (h@W2)
    long long t16 = (long long)E * 16;
    k_spmm16<<<(unsigned)((t16 + 255) / 256), 256, 0, stream>>>(rows, cols, vals,
                                                                hw2_ws, acc2_ws, E);

    // + b2, log_softmax
    k_lsm<<<(N + 255) / 256, 256, 0, stream>>>(acc2_ws, b2, out, N);
}